// CropProposals_57251914056329
// MI455X (gfx1250) — compile-verified
//
#include <hip/hip_runtime.h>
#include <math.h>

// 3D ROI max-pool (2x2x2 bins), proposals x channels crossed.
// fm: (4,64,24,24,24) f32, corners: (4,64,2,3) f32 -> out: (4,64,64,2,2,2) f32.
//
// Design: one wave32 per (b,p,c). Lanes = W axis (stride-1, coalesced,
// predicated to the exact W region). D/H bin tests are wave-uniform scalars.
// kz split resolved by masked wave32 butterfly max-reductions (__shfl_xor).
// global_prefetch_b8 hides the D-plane stride. WMMA is inapplicable (max, not
// multiply-accumulate); this op is L2-latency bound, not compute bound.

#define NEG_INF (-__builtin_huge_valf())

__device__ __forceinline__ float wave_max32(float v) {
#pragma unroll
  for (int m = 16; m >= 1; m >>= 1)
    v = fmaxf(v, __shfl_xor(v, m, 32));
  return v;
}

__global__ __launch_bounds__(256) void roi_maxpool3d_kernel(
    const float* __restrict__ fm, const float* __restrict__ corners,
    float* __restrict__ out) {
  const int lane = threadIdx.x & 31;
  const int gw   = blockIdx.x * 8 + (threadIdx.x >> 5);  // global wave id
  const int b    = gw >> 12;           // 4 batches
  const int p    = (gw >> 6) & 63;     // 64 proposals
  const int c    = gw & 63;            // 64 channels

  // ---- per-(b,p) bin boundaries (wave-uniform), exact reference semantics ----
  const float* cr = corners + ((b << 6) + p) * 6;
  int lo[3], n[3];
#pragma unroll
  for (int a = 0; a < 3; ++a) {
    float LL = fminf(fmaxf(cr[a] * 0.25f, 0.0f), 21.0f);
    float UR = cr[3 + a] * 0.25f;
    UR = (UR - LL >= 2.0f) ? UR : (LL + 2.0f);
    UR = fminf(fmaxf(UR, 2.0f), 23.0f);
    lo[a] = (int)floorf(LL);
    n[a]  = (int)floorf(UR) - lo[a];   // n in [2, 23]
  }
  // bin k=0: [lo, lo+(n+1)/2)   bin k=1: [lo+n/2, lo+n)
  const int xlo = lo[0], xn = n[0], xhi0 = xlo + ((xn + 1) >> 1), xlo1 = xlo + (xn >> 1);
  const int ylo = lo[1], yn = n[1], yhi0 = ylo + ((yn + 1) >> 1), ylo1 = ylo + (yn >> 1);
  const int zlo = lo[2], zn = n[2], zhi0 = zlo + ((zn + 1) >> 1), zlo1 = zlo + (zn >> 1);

  const float* base = fm + (size_t)((b << 6) + c) * 13824u;  // 24*24*24

  // Lane -> W position; only lanes inside the W region load (exact-line fetch).
  const bool inW = (lane >= zlo) & (lane < zlo + zn);

  float a00 = NEG_INF, a01 = NEG_INF, a10 = NEG_INF, a11 = NEG_INF;

  for (int x = xlo; x < xlo + xn; ++x) {
    const bool fx0 = (x < xhi0);
    const bool fx1 = (x >= xlo1);
    const float* px = base + x * 576;
    // Prefetch next D-plane row range (gfx1250 global_prefetch_b8).
    __builtin_prefetch(px + 576 + ylo * 24 + lane, 0, 0);
    for (int y = ylo; y < ylo + yn; ++y) {
      const float v   = inW ? px[y * 24 + lane] : NEG_INF;
      const bool  fy0 = (y < yhi0);
      const bool  fy1 = (y >= ylo1);
      a00 = fmaxf(a00, (fx0 & fy0) ? v : NEG_INF);
      a01 = fmaxf(a01, (fx0 & fy1) ? v : NEG_INF);
      a10 = fmaxf(a10, (fx1 & fy0) ? v : NEG_INF);
      a11 = fmaxf(a11, (fx1 & fy1) ? v : NEG_INF);
    }
  }

  // kz split: per-lane W membership in each z-bin, then wave32 max-butterflies.
  const bool fz0 = (lane >= zlo)  & (lane < zhi0);
  const bool fz1 = (lane >= zlo1) & (lane < zlo + zn);

  float r0 = wave_max32(fz0 ? a00 : NEG_INF);  // (kx=0,ky=0,kz=0)
  float r1 = wave_max32(fz1 ? a00 : NEG_INF);  // (0,0,1)
  float r2 = wave_max32(fz0 ? a01 : NEG_INF);  // (0,1,0)
  float r3 = wave_max32(fz1 ? a01 : NEG_INF);  // (0,1,1)
  float r4 = wave_max32(fz0 ? a10 : NEG_INF);  // (1,0,0)
  float r5 = wave_max32(fz1 ? a10 : NEG_INF);  // (1,0,1)
  float r6 = wave_max32(fz0 ? a11 : NEG_INF);  // (1,1,0)
  float r7 = wave_max32(fz1 ? a11 : NEG_INF);  // (1,1,1)

  if (lane == 0) {
    float4* o = (float4*)(out + (size_t)gw * 8);  // 32B-aligned
    o[0] = make_float4(r0, r1, r2, r3);
    o[1] = make_float4(r4, r5, r6, r7);
  }
}

extern "C" void kernel_launch(void* const* d_in, const int* in_sizes, int n_in,
                              void* d_out, int out_size, void* d_ws, size_t ws_size,
                              hipStream_t stream) {
  (void)in_sizes; (void)n_in; (void)d_ws; (void)ws_size; (void)out_size;
  const float* fm      = (const float*)d_in[0];
  const float* corners = (const float*)d_in[1];
  float*       out     = (float*)d_out;
  // 4*64*64 = 16384 waves = 2048 blocks x 8 waves (256 threads, wave32).
  hipLaunchKernelGGL(roi_maxpool3d_kernel, dim3(2048), dim3(256), 0, stream,
                     fm, corners, out);
}